// regressor_70669391889130
// MI455X (gfx1250) — compile-verified
//
#include <hip/hip_runtime.h>
#include <hip/hip_bf16.h>
#include <math.h>

typedef __attribute__((ext_vector_type(2))) float v2f;
typedef __attribute__((ext_vector_type(8))) float v8f;
typedef __attribute__((address_space(1))) int as1_i32;
typedef __attribute__((address_space(3))) int as3_i32;

#define EPS_STD 0.001f
constexpr int B = 4, N = 4096, Q = 2048, M = 2048;
constexpr int KB = 16;  // K-block staged in LDS (4 wmma k-steps)

#if __has_builtin(__builtin_amdgcn_global_load_async_to_lds_b32) && \
    __has_builtin(__builtin_amdgcn_s_wait_asynccnt)
#define USE_ASYNC_LDS 1
#else
#define USE_ASYNC_LDS 0
#endif

// One DWORD per lane: global -> LDS, tracked by ASYNCcnt (no VGPR staging).
__device__ __forceinline__ void async_copy_b32(float* lds_dst, const float* gsrc) {
#if USE_ASYNC_LDS
  __builtin_amdgcn_global_load_async_to_lds_b32(
      (as1_i32*)(uintptr_t)gsrc,
      (as3_i32*)(uint32_t)(uintptr_t)lds_dst,  // AS3 = low 32 bits (LDS offset)
      0, 0);
#else
  *lds_dst = *gsrc;
#endif
}

__device__ __forceinline__ void wait_async_le16() {
#if USE_ASYNC_LDS
  __builtin_amdgcn_s_wait_asynccnt(16);
#endif
}
__device__ __forceinline__ void wait_async_le0() {
#if USE_ASYNC_LDS
  __builtin_amdgcn_s_wait_asynccnt(0);
#endif
}

__device__ __forceinline__ v8f wmma4(v2f a, v2f b, v8f c) {
  // V_WMMA_F32_16X16X4_F32 : D = A(16x4) x B(4x16) + C(16x16), all f32
  return __builtin_amdgcn_wmma_f32_16x16x4_f32(
      false, a, false, b, (short)0, c, false, false);
}

// ---------------- column stats of Y: mu, s=std(ddof=1)+eps, inv_s, yy=sum(Yn^2) ---
__global__ __launch_bounds__(256) void stats_y_kernel(
    const float* __restrict__ Yg, float* __restrict__ mu, float* __restrict__ s,
    float* __restrict__ inv_s, float* __restrict__ yy) {
  int tid = blockIdx.x * blockDim.x + threadIdx.x;  // 0 .. B*M-1
  int b = tid / M, m = tid % M;
  const float* col = Yg + (size_t)b * N * M + m;
  float sum = 0.f, sq = 0.f;
  for (int n = 0; n < N; ++n) {
    float v = col[(size_t)n * M];
    sum += v; sq += v * v;
  }
  float mean = sum * (1.0f / N);
  float ss = fmaxf(sq - (float)N * mean * mean, 0.f);  // sum of centered squares
  float var = ss * (1.0f / (N - 1));                   // ddof=1
  float sd = sqrtf(var) + EPS_STD;
  mu[tid] = mean;
  s[tid] = sd;
  inv_s[tid] = 1.0f / sd;
  yy[tid] = ss / (sd * sd);
}

// ---------------- column stats of X: sx = sum, xx = sum of squares ---------------
__global__ __launch_bounds__(256) void stats_x_kernel(
    const float* __restrict__ Xg, float* __restrict__ sx, float* __restrict__ xx) {
  int tid = blockIdx.x * blockDim.x + threadIdx.x;  // 0 .. B*Q-1
  int b = tid / Q, q = tid % Q;
  const float* col = Xg + (size_t)b * N * Q + q;
  float sum = 0.f, sq = 0.f;
  for (int n = 0; n < N; ++n) {
    float v = col[(size_t)n * Q];
    sum += v; sq += v * v;
  }
  sx[tid] = sum;
  xx[tid] = sq;
}

// ---------------- GEMM1: l2[q,m] = -sqrt(clip(xx[q]+yy[m]-2*xy, 0)) ---------------
// xy = (sum_n X[n,q]*Y[n,m] - sx[q]*mu[m]) * inv_s[m]   (avoids materializing Yn)
// 128x128 block tile, double-buffered LDS filled with async-to-LDS copies.
__global__ __launch_bounds__(256) void gemm1_kernel(
    const float* __restrict__ Xg, const float* __restrict__ Yg,
    const float* __restrict__ sx, const float* __restrict__ xxv,
    const float* __restrict__ mu, const float* __restrict__ inv_s,
    const float* __restrict__ yyv, float* __restrict__ P) {
  __shared__ alignas(16) float As[2][128][18];  // [buf][q-local][k], 8B-aligned v2f
  __shared__ alignas(16) float Bs[2][16][130];  // [buf][k][m-local]

  const int b = blockIdx.z;
  const int m0 = blockIdx.x * 128;
  const int q0 = blockIdx.y * 128;
  const float* Xb = Xg + (size_t)b * N * Q;
  const float* Yb = Yg + (size_t)b * N * M;

  const int t = threadIdx.x;
  const int lane = t & 31, wid = t >> 5;
  const int wm = wid & 3, wn = wid >> 2;  // wave grid 4x2: 32-row x 64-col patches
  const int h = lane >> 4, ln = lane & 15;

  v8f acc[2][4] = {};

  auto issue = [&](int buf, int kb) {  // 16 async b32 copies per thread
#pragma unroll
    for (int i = 0; i < 8; ++i) {  // A: As[c][r] = X[(kb+r)*Q + q0+c]
      int idx = t + i * 256;
      int r = idx >> 7, c = idx & 127;
      async_copy_b32(&As[buf][c][r], Xb + (size_t)(kb + r) * Q + q0 + c);
    }
#pragma unroll
    for (int i = 0; i < 8; ++i) {  // B: Bs[r][c] = Y[(kb+r)*M + m0+c]
      int idx = t + i * 256;
      int r = idx >> 7, c = idx & 127;
      async_copy_b32(&Bs[buf][r][c], Yb + (size_t)(kb + r) * M + m0 + c);
    }
  };

  issue(0, 0);
  for (int kb = 0; kb < N; kb += KB) {
    const int cur = (kb / KB) & 1;
    if (kb + KB < N) {
      issue(cur ^ 1, kb + KB);
      wait_async_le16();  // current buffer's 16 loads retired (in-order)
    } else {
      wait_async_le0();
    }
    __syncthreads();
#pragma unroll
    for (int k4 = 0; k4 < KB; k4 += 4) {
      const int ka = k4 + 2 * h;  // ISA A/B frag: K = 2*(lane>>4) + v
      v2f a0 = *(const v2f*)&As[cur][wm * 32 + ln][ka];
      v2f a1 = *(const v2f*)&As[cur][wm * 32 + 16 + ln][ka];
      v2f bf[4];
#pragma unroll
      for (int j = 0; j < 4; ++j) {
        bf[j].x = Bs[cur][ka][wn * 64 + j * 16 + ln];
        bf[j].y = Bs[cur][ka + 1][wn * 64 + j * 16 + ln];
      }
#pragma unroll
      for (int j = 0; j < 4; ++j) {
        acc[0][j] = wmma4(a0, bf[j], acc[0][j]);
        acc[1][j] = wmma4(a1, bf[j], acc[1][j]);
      }
    }
    __syncthreads();
  }

  const float* sxb = sx + b * Q;
  const float* xxb = xxv + b * Q;
  const float* mub = mu + b * M;
  const float* isb = inv_s + b * M;
  const float* yyb = yyv + b * M;
  float* Pb = P + (size_t)b * Q * M;
#pragma unroll
  for (int i = 0; i < 2; ++i)
#pragma unroll
    for (int j = 0; j < 4; ++j) {
      const int col = m0 + wn * 64 + j * 16 + ln;
      const float muv = mub[col], isv = isb[col], yv = yyb[col];
#pragma unroll
      for (int r = 0; r < 8; ++r) {  // C/D layout: M = r + 8*(lane>>4), N = lane&15
        const int row = q0 + wm * 32 + i * 16 + r + 8 * h;
        const float raw = acc[i][j][r];
        const float xy = (raw - sxb[row] * muv) * isv;
        const float d2 = fmaxf(xxb[row] + yv - 2.0f * xy, 0.0f);
        Pb[(size_t)row * M + col] = -sqrtf(d2);
      }
    }
}

// ---------------- softmax over Q per (b,m) column; leaves unnormalized exp in P ---
// Normalization 1/sum is folded into GEMM2 epilogue.
__global__ __launch_bounds__(256) void softmax_col_kernel(
    float* __restrict__ P, float* __restrict__ inv_sum) {
  __shared__ float red[4][64];
  const int b = blockIdx.y;
  const int tx = threadIdx.x, ty = threadIdx.y;
  const int m = blockIdx.x * 64 + tx;
  float* Pb = P + (size_t)b * Q * M + m;

  float lmax = -3.4e38f;
  for (int q = ty; q < Q; q += 4) lmax = fmaxf(lmax, Pb[(size_t)q * M]);
  red[ty][tx] = lmax;
  __syncthreads();
  const float cmax = fmaxf(fmaxf(red[0][tx], red[1][tx]),
                           fmaxf(red[2][tx], red[3][tx]));
  __syncthreads();

  float lsum = 0.f;
  for (int q = ty; q < Q; q += 4) {
    float e = __expf(Pb[(size_t)q * M] - cmax);
    Pb[(size_t)q * M] = e;
    lsum += e;
  }
  red[ty][tx] = lsum;
  __syncthreads();
  if (ty == 0) {
    float sum = red[0][tx] + red[1][tx] + red[2][tx] + red[3][tx];
    inv_sum[b * M + m] = 1.0f / sum;
  }
}

// ---------------- GEMM2: k = X@p, kstd = (X*X)@p fused; y/ystd epilogue ----------
__global__ __launch_bounds__(256) void gemm2_kernel(
    const float* __restrict__ Xg, const float* __restrict__ P,
    const float* __restrict__ mu, const float* __restrict__ s,
    const float* __restrict__ inv_sum, float* __restrict__ yo,
    float* __restrict__ ystd) {
  __shared__ alignas(16) float As[2][128][18];  // [buf][n-local][k(=q)]
  __shared__ alignas(16) float Bs[2][16][130];  // [buf][k][m-local]

  const int b = blockIdx.z;
  const int m0 = blockIdx.x * 128;
  const int n0 = blockIdx.y * 128;
  const float* Xb = Xg + (size_t)b * N * Q;
  const float* Pb = P + (size_t)b * Q * M;

  const int t = threadIdx.x;
  const int lane = t & 31, wid = t >> 5;
  const int wm = wid & 3, wn = wid >> 2;
  const int h = lane >> 4, ln = lane & 15;

  v8f ak[2][4] = {};  // k accumulator
  v8f a2[2][4] = {};  // kstd accumulator

  auto issue = [&](int buf, int kb) {  // 16 async b32 copies per thread
#pragma unroll
    for (int i = 0; i < 8; ++i) {  // A: As[r][c] = X[(n0+r)*Q + kb+c]
      int idx = t + i * 256;
      int r = idx >> 4, c = idx & 15;
      async_copy_b32(&As[buf][r][c], Xb + (size_t)(n0 + r) * Q + kb + c);
    }
#pragma unroll
    for (int i = 0; i < 8; ++i) {  // B: Bs[r][c] = P[(kb+r)*M + m0+c]
      int idx = t + i * 256;
      int r = idx >> 7, c = idx & 127;
      async_copy_b32(&Bs[buf][r][c], Pb + (size_t)(kb + r) * M + m0 + c);
    }
  };

  issue(0, 0);
  for (int kb = 0; kb < Q; kb += KB) {
    const int cur = (kb / KB) & 1;
    if (kb + KB < Q) {
      issue(cur ^ 1, kb + KB);
      wait_async_le16();
    } else {
      wait_async_le0();
    }
    __syncthreads();
#pragma unroll
    for (int k4 = 0; k4 < KB; k4 += 4) {
      const int ka = k4 + 2 * h;
      v2f x0 = *(const v2f*)&As[cur][wm * 32 + ln][ka];
      v2f x1 = *(const v2f*)&As[cur][wm * 32 + 16 + ln][ka];
      v2f x0s = x0 * x0;  // X^2 fragment in registers: X read once
      v2f x1s = x1 * x1;
      v2f bf[4];
#pragma unroll
      for (int j = 0; j < 4; ++j) {
        bf[j].x = Bs[cur][ka][wn * 64 + j * 16 + ln];
        bf[j].y = Bs[cur][ka + 1][wn * 64 + j * 16 + ln];
      }
#pragma unroll
      for (int j = 0; j < 4; ++j) {
        ak[0][j] = wmma4(x0, bf[j], ak[0][j]);
        ak[1][j] = wmma4(x1, bf[j], ak[1][j]);
        a2[0][j] = wmma4(x0s, bf[j], a2[0][j]);
        a2[1][j] = wmma4(x1s, bf[j], a2[1][j]);
      }
    }
    __syncthreads();
  }

  const float* mub = mu + b * M;
  const float* sb = s + b * M;
  const float* ivb = inv_sum + b * M;
  float* yb = yo + (size_t)b * N * M;
  float* eb = ystd + (size_t)b * N * M;
#pragma unroll
  for (int i = 0; i < 2; ++i)
#pragma unroll
    for (int j = 0; j < 4; ++j) {
      const int col = m0 + wn * 64 + j * 16 + ln;
      const float muv = mub[col], sv = sb[col], iv = ivb[col];
#pragma unroll
      for (int r = 0; r < 8; ++r) {
        const int row = n0 + wm * 32 + i * 16 + r + 8 * h;
        const float kk = ak[i][j][r] * iv;  // softmax normalization folded here
        const float ks = a2[i][j][r] * iv;
        yb[(size_t)row * M + col] = kk * sv + muv;
        eb[(size_t)row * M + col] = sqrtf(fmaxf(ks - kk * kk, 0.0f)) * sv;
      }
    }
}

extern "C" void kernel_launch(void* const* d_in, const int* in_sizes, int n_in,
                              void* d_out, int out_size, void* d_ws, size_t ws_size,
                              hipStream_t stream) {
  (void)in_sizes; (void)n_in; (void)out_size; (void)ws_size;
  const float* X = (const float*)d_in[0];
  const float* Y = (const float*)d_in[1];

  float* ws = (float*)d_ws;
  float* mu  = ws;           // B*M
  float* s   = ws + 8192;    // B*M
  float* isd = ws + 16384;   // B*M  (1/s)
  float* yy  = ws + 24576;   // B*M
  float* sx  = ws + 32768;   // B*Q
  float* xx  = ws + 40960;   // B*Q
  float* ivs = ws + 49152;   // B*M  (1/sum exp)
  float* P   = ws + 57344;   // B*Q*M  (l2 -> exp buffer, 64 MiB)

  float* y_out = (float*)d_out;
  float* ystd  = y_out + (size_t)B * N * M;

  stats_y_kernel<<<(B * M) / 256, 256, 0, stream>>>(Y, mu, s, isd, yy);
  stats_x_kernel<<<(B * Q) / 256, 256, 0, stream>>>(X, sx, xx);
  gemm1_kernel<<<dim3(M / 128, Q / 128, B), 256, 0, stream>>>(
      X, Y, sx, xx, mu, isd, yy, P);
  softmax_col_kernel<<<dim3(M / 64, B), dim3(64, 4), 0, stream>>>(P, ivs);
  gemm2_kernel<<<dim3(M / 128, N / 128, B), 256, 0, stream>>>(
      X, P, mu, s, ivs, y_out, ystd);
}